// ParameterLayerBase_13211319402579
// MI455X (gfx1250) — compile-verified
//
#include <hip/hip_runtime.h>
#include <hip/hip_bf16.h>

// Problem sizes (fixed by the reference)
#define B_SZ   128
#define IN_SZ  1024
#define OUT_SZ 1024
#define E_SZ   32

typedef __attribute__((ext_vector_type(2))) float v2f;
typedef __attribute__((ext_vector_type(8))) float v8f;

// ---------------------------------------------------------------------------
// Wave32 top-2 (value, index) butterfly reduction. Tie-break: lower index
// (matches jax.lax.top_k first-occurrence semantics). All lanes converge to
// the same result.
// ---------------------------------------------------------------------------
__device__ inline void wave_top2(float val, int lane,
                                 int& i0, int& i1, float& v0, float& v1) {
  float v = val; int idx = lane;
#pragma unroll
  for (int off = 16; off > 0; off >>= 1) {
    float ov = __shfl_xor(v, off, 32);
    int   oi = __shfl_xor(idx, off, 32);
    if (ov > v || (ov == v && oi < idx)) { v = ov; idx = oi; }
  }
  v0 = v; i0 = idx;
  float v2 = (lane == i0) ? -__builtin_inff() : val;
  float vv = v2; int idx2 = lane;
#pragma unroll
  for (int off = 16; off > 0; off >>= 1) {
    float ov = __shfl_xor(vv, off, 32);
    int   oi = __shfl_xor(idx2, off, 32);
    if (ov > vv || (ov == vv && oi < idx2)) { vv = ov; idx2 = oi; }
  }
  v1 = vv; i1 = idx2;
}

// ---------------------------------------------------------------------------
// Kernel 1: routing. One wave32 per sample.
//  - lane e computes logits[b][e] for both routers (E=32 == wave width)
//  - top-2 + softmax; token grouped into its 2 experts' lists (d_ws)
//  - bias mixture written directly to out[b][:], initializing the output
// ---------------------------------------------------------------------------
__global__ __launch_bounds__(32) void moe_route_kernel(
    const float* __restrict__ x, const float* __restrict__ rw,
    const float* __restrict__ brw, const float* __restrict__ eb,
    float* __restrict__ out, int* __restrict__ cnt,
    int* __restrict__ gtok, float* __restrict__ gprob) {
  const int b = blockIdx.x;
  const int lane = threadIdx.x;
  const float* xb = x + (size_t)b * IN_SZ;

  float accr = 0.f, accb = 0.f;
#pragma unroll 4
  for (int i = 0; i < IN_SZ; ++i) {
    const float xv = xb[i];                      // wave-uniform (scalarized)
    accr = __builtin_fmaf(xv, rw[i * E_SZ + lane], accr);   // coalesced
    accb = __builtin_fmaf(xv, brw[i * E_SZ + lane], accb);  // coalesced
  }

  // Weight-router top-2 -> expert grouping
  int i0, i1; float v0, v1;
  wave_top2(accr, lane, i0, i1, v0, v1);
  const float ex = __expf(v1 - v0);              // v1 <= v0
  const float inv = 1.f / (1.f + ex);
  const float p0 = inv, p1 = ex * inv;
  if (lane == 0) {
    int s0 = atomicAdd(&cnt[i0], 1);
    gtok[i0 * B_SZ + s0] = b; gprob[i0 * B_SZ + s0] = p0;
    int s1 = atomicAdd(&cnt[i1], 1);
    gtok[i1 * B_SZ + s1] = b; gprob[i1 * B_SZ + s1] = p1;
  }

  // Bias-router top-2 -> write generated bias into out (initializes d_out)
  int j0, j1; float u0, u1;
  wave_top2(accb, lane, j0, j1, u0, u1);
  const float exb = __expf(u1 - u0);
  const float invb = 1.f / (1.f + exb);
  const float q0 = invb, q1 = exb * invb;
  const float* b0p = eb + (size_t)j0 * OUT_SZ;
  const float* b1p = eb + (size_t)j1 * OUT_SZ;
  float* ob = out + (size_t)b * OUT_SZ;
#pragma unroll 4
  for (int o = lane; o < OUT_SZ; o += 32)
    ob[o] = __builtin_fmaf(q0, b0p[o], q1 * b1p[o]);
}

// ---------------------------------------------------------------------------
// Kernel 2: grouped expert GEMM with f32 WMMA.
// grid = (OUT/128, E, B/16). Block = 256 threads = 8 waves.
// Each block: one expert, one 128-column tile, one chunk of <=16 grouped
// tokens. A (16 x 1024 token rows) staged through LDS in K-chunks of 256;
// each wave owns a 16-col subtile and iterates v_wmma_f32_16x16x4_f32.
// W is streamed coalesced from HBM exactly once per (expert, col-tile).
// ---------------------------------------------------------------------------
#define KCHUNK 256
#define APITCH 260   // 260 % 64 == 4 -> row r lands on bank 4r: conflict-free

__global__ __launch_bounds__(256) void moe_gemm_kernel(
    const float* __restrict__ x, const float* __restrict__ ew,
    float* __restrict__ out, const int* __restrict__ cnt,
    const int* __restrict__ gtok, const float* __restrict__ gprob) {
  __shared__ float sA[16][APITCH];
  __shared__ int   s_tok[16];
  __shared__ float s_prob[16];

  const int e     = blockIdx.y;
  const int mbase = blockIdx.z * 16;
  const int cnt_e = cnt[e];
  if (mbase >= cnt_e) return;                    // block-uniform exit
  const int rows = min(16, cnt_e - mbase);

  const int tid = threadIdx.x;
  if (tid < 16) {
    if (tid < rows) {
      s_tok[tid]  = gtok[e * B_SZ + mbase + tid];
      s_prob[tid] = gprob[e * B_SZ + mbase + tid];
    } else { s_tok[tid] = 0; s_prob[tid] = 0.f; }
  }
  __syncthreads();

  const int wave = tid >> 5;
  const int lane = tid & 31;
  const int half = lane >> 4;                    // 0: K+{0,1}; 1: K+{2,3}
  const int l16  = lane & 15;
  const int col  = blockIdx.x * 128 + wave * 16 + l16;
  const float* Wp = ew + (size_t)e * IN_SZ * OUT_SZ;

  // Cooperative A-load geometry: 256 threads x 16 floats = 16 x 256 chunk
  const int arow = tid >> 4;                     // 0..15
  const int aseg = (tid & 15) * 16;              // 0..240 step 16
  const int atok = s_tok[arow];                  // valid after barrier

  v8f acc = {};

  for (int kb = 0; kb < IN_SZ; kb += KCHUNK) {
    // Stage x rows of this token group into LDS (float4, coalesced; zero-pad)
    {
      float4* dst = (float4*)&sA[arow][aseg];
      if (arow < rows) {
        const float4* src = (const float4*)(x + (size_t)atok * IN_SZ + kb + aseg);
#pragma unroll
        for (int j = 0; j < 4; ++j) dst[j] = src[j];
      } else {
        const float4 z = make_float4(0.f, 0.f, 0.f, 0.f);
#pragma unroll
        for (int j = 0; j < 4; ++j) dst[j] = z;
      }
    }
    __syncthreads();

    // Prefetch next K-chunk of W for this column stream
    if (kb + KCHUNK < IN_SZ)
      __builtin_prefetch(&Wp[(size_t)(kb + KCHUNK + 2 * half) * OUT_SZ + col], 0, 1);

#pragma unroll 8
    for (int k4 = 0; k4 < KCHUNK; k4 += 4) {
      // A fragment (16x4 f32): lane holds row l16, K = 2*half + {0,1}
      const float2 av = *(const float2*)&sA[l16][k4 + 2 * half];
      v2f a; a[0] = av.x; a[1] = av.y;
      // B fragment (4x16 f32): lane holds col, rows K = 2*half + {0,1}
      const size_t krow = (size_t)(kb + k4 + 2 * half) * OUT_SZ + col;
      v2f bb; bb[0] = Wp[krow]; bb[1] = Wp[krow + OUT_SZ];
      acc = __builtin_amdgcn_wmma_f32_16x16x4_f32(
          /*neg_a=*/false, a, /*neg_b=*/false, bb,
          /*c_mod=*/(short)0, acc, /*reuse_a=*/false, /*reuse_b=*/false);
    }
    __syncthreads();
  }

  // Scale rows by routing prob, scatter-add into out[token][col].
  // Each out element gets exactly 2 atomic f32 adds (commutative) on top of
  // the bias written by kernel 1 -> add order varies, so not bitwise deterministic.
#pragma unroll
  for (int v = 0; v < 8; ++v) {
    const int r = half ? (v + 8) : v;            // C/D layout: lanes>=16 -> M+8
    if (r < rows)
      atomicAdd(&out[(size_t)s_tok[r] * OUT_SZ + col], s_prob[r] * acc[v]);
  }
}

// ---------------------------------------------------------------------------
// Host launcher
// ---------------------------------------------------------------------------
extern "C" void kernel_launch(void* const* d_in, const int* in_sizes, int n_in,
                              void* d_out, int out_size, void* d_ws, size_t ws_size,
                              hipStream_t stream) {
  const float* x   = (const float*)d_in[0];
  const float* rw  = (const float*)d_in[1];
  const float* brw = (const float*)d_in[2];
  const float* ew  = (const float*)d_in[3];
  const float* eb  = (const float*)d_in[4];
  float* out = (float*)d_out;

  // Workspace layout: [cnt: E ints (padded to 256B)] [gtok: E*B ints] [gprob: E*B floats]
  int*   cnt   = (int*)d_ws;
  int*   gtok  = (int*)((char*)d_ws + 256);
  float* gprob = (float*)((char*)d_ws + 256 + (size_t)E_SZ * B_SZ * sizeof(int));

  hipMemsetAsync(cnt, 0, E_SZ * sizeof(int), stream);
  moe_route_kernel<<<B_SZ, 32, 0, stream>>>(x, rw, brw, eb, out, cnt, gtok, gprob);
  dim3 g2(OUT_SZ / 128, E_SZ, B_SZ / 16);
  moe_gemm_kernel<<<g2, 256, 0, stream>>>(x, ew, out, cnt, gtok, gprob);
}